// SelfConditionPairwiseAtomicEmbedding_56023553409080
// MI455X (gfx1250) — compile-verified
//
#include <hip/hip_runtime.h>

#define SCALE   0.1f
#define NUM_AA  21
#define NUM_RBF 64
#define EPS     1e-8f
#define NCOLS   387          // 1 + 21 + 21 + 64 + 42 + 42 + 196
#define EPB     8            // edges (waves) per 256-thread block

typedef __attribute__((ext_vector_type(2))) float v2f;
typedef __attribute__((ext_vector_type(8))) float v8f;

// ---------------------------------------------------------------------------
// Phase 1: per-node precompute.
//   seqArr[n]     = argmax(logits[n])            (first-max wins, like jnp)
//   nodeA[n][a]   = (l0, l1, l2, |l+EPS|^2)      a = 0..13 ; rows 14,15 = 0
//   where l = rot[n]^T (SCALE*atom14[n,a] - SCALE*trans[n])
// ---------------------------------------------------------------------------
__global__ void node_pre_kernel(const float* __restrict__ atom14,
                                const float* __restrict__ rot,
                                const float* __restrict__ trans,
                                const float* __restrict__ logits,
                                float4* __restrict__ nodeA,
                                int* __restrict__ seqArr,
                                int N)
{
    int n = blockIdx.x * blockDim.x + threadIdx.x;
    if (n >= N) return;

    const float* lg = logits + (size_t)n * NUM_AA;
    int best = 0; float bv = lg[0];
    #pragma unroll
    for (int k = 1; k < NUM_AA; ++k) { float v = lg[k]; if (v > bv) { bv = v; best = k; } }
    seqArr[n] = best;

    float R[9];
    #pragma unroll
    for (int i = 0; i < 9; ++i) R[i] = rot[(size_t)n * 9 + i];
    float t0 = SCALE * trans[(size_t)n * 3 + 0];
    float t1 = SCALE * trans[(size_t)n * 3 + 1];
    float t2 = SCALE * trans[(size_t)n * 3 + 2];

    const float* at = atom14 + (size_t)n * 42;
    #pragma unroll
    for (int a = 0; a < 14; ++a) {
        float v0 = SCALE * at[a*3+0] - t0;
        float v1 = SCALE * at[a*3+1] - t1;
        float v2 = SCALE * at[a*3+2] - t2;
        float l0 = R[0]*v0 + R[3]*v1 + R[6]*v2;   // out_i = sum_j R[j*3+i] v_j
        float l1 = R[1]*v0 + R[4]*v1 + R[7]*v2;
        float l2 = R[2]*v0 + R[5]*v1 + R[8]*v2;
        float a0 = l0 + EPS, a1 = l1 + EPS, a2 = l2 + EPS;
        nodeA[(size_t)n*16 + a] = make_float4(l0, l1, l2, a0*a0 + a1*a1 + a2*a2);
    }
    nodeA[(size_t)n*16 + 14] = make_float4(0.f, 0.f, 0.f, 0.f);
    nodeA[(size_t)n*16 + 15] = make_float4(0.f, 0.f, 0.f, 0.f);
}

// ---------------------------------------------------------------------------
// Phase 2: one wave per edge. A single V_WMMA_F32_16X16X4_F32 computes
//   D[m][n] = |a'_m|^2 - 2 a'_m . b_n        (|a'|^2 folded into K=4 slot)
// so pair_dist = sqrt(max(D + |b_n|^2, 0)).  387-column row is staged in LDS
// and flushed with coalesced lane-strided stores.
// ---------------------------------------------------------------------------
__global__ __launch_bounds__(256) void edge_kernel(
        const float* __restrict__ atom14,
        const float* __restrict__ trans,
        const int*   __restrict__ eidx,
        const float4* __restrict__ nodeA,
        const int*   __restrict__ seqArr,
        const float* __restrict__ rot,
        float* __restrict__ out,
        int E)
{
    __shared__ float rows[EPB][NCOLS + 1];     // slot NCOLS = dump pad

    const int lane = threadIdx.x & 31;
    const int wv   = threadIdx.x >> 5;
    const int e    = blockIdx.x * EPB + wv;
    if (e >= E) return;                        // wave-uniform branch

    float* row = rows[wv];

    // Edge endpoints, forced wave-uniform -> scalar loads for node data.
    int dst = __builtin_amdgcn_readfirstlane(eidx[(size_t)e]);
    int src = __builtin_amdgcn_readfirstlane(eidx[(size_t)E + (size_t)e]);

    // ---- uniform node scalars -------------------------------------------
    const float* Rs = rot + (size_t)src * 9;
    float r0=Rs[0], r1=Rs[1], r2=Rs[2], r3=Rs[3], r4=Rs[4],
          r5=Rs[5], r6=Rs[6], r7=Rs[7], r8=Rs[8];
    float tx = SCALE * trans[(size_t)src*3 + 0];
    float ty = SCALE * trans[(size_t)src*3 + 1];
    float tz = SCALE * trans[(size_t)src*3 + 2];
    int   ss = seqArr[src];
    int   sd = seqArr[dst];

    // ---- b_j = rot[src]^T (SCALE*atom14[dst,j] - t[src]); lane j = atom j
    float b0 = 0.f, b1 = 0.f, b2 = 0.f, bsq = 0.f;
    if (lane < 14) {
        const float* ad = atom14 + (size_t)dst*42 + lane*3;
        float v0 = SCALE*ad[0] - tx;
        float v1 = SCALE*ad[1] - ty;
        float v2 = SCALE*ad[2] - tz;
        b0 = r0*v0 + r3*v1 + r6*v2;
        b1 = r1*v0 + r4*v1 + r7*v2;
        b2 = r2*v0 + r5*v1 + r8*v2;
        bsq = b0*b0 + b1*b1 + b2*b2;
    }

    // ---- WMMA operands ---------------------------------------------------
    const bool lo = (lane < 16);
    const int  m  = lane & 15;                 // A row this lane carries
    float4 Am = nodeA[(size_t)src*16 + m];     // (local xyz, |local+EPS|^2)
    float ax = Am.x + EPS, ay = Am.y + EPS, az = Am.z + EPS;

    // A (16x4): rows m = a'; K slots = (-2a0, -2a1, -2a2, |a'|^2)
    v2f av;
    av[0] = lo ? (-2.0f * ax) : (-2.0f * az);
    av[1] = lo ? (-2.0f * ay) : Am.w;

    // B (4x16): cols n = b;  K slots = (b0, b1, b2, 1)
    const int nn = lane & 15;
    float bn0 = __shfl(b0, nn, 32);
    float bn1 = __shfl(b1, nn, 32);
    float bn2 = __shfl(b2, nn, 32);
    v2f bvv;
    bvv[0] = lo ? bn0 : bn2;
    bvv[1] = lo ? bn1 : 1.0f;

    v8f acc = {};                              // C = 0
    v8f dmat = __builtin_amdgcn_wmma_f32_16x16x4_f32(
        false, av, false, bvv, (short)0, acc, false, false);

    // ---- pair distances (branch-free; invalid lanes hit the dump slot) --
    float bsqn = __shfl(bsq, nn, 32);
    const int mbase = lo ? 0 : 8;
    const bool ncol_ok = (nn < 14);
    #pragma unroll
    for (int r = 0; r < 8; ++r) {
        int   mrow = r + mbase;
        float dist = __builtin_amdgcn_sqrtf(fmaxf(dmat[r] + bsqn, 0.0f));
        bool  ok   = ncol_ok && (mrow < 14);
        int   idx  = ok ? (191 + mrow*14 + nn) : NCOLS;
        row[idx] = dist;
    }

    // ---- remaining columns ----------------------------------------------
    if (lane == 0) row[0] = 1.0f;
    if (lane < NUM_AA) {
        row[1  + lane] = (lane == ss) ? 1.0f : 0.0f;
        row[22 + lane] = (lane == sd) ? 1.0f : 0.0f;
    }

    // Calpha distance (uniform across the wave)
    float dx = SCALE * (atom14[(size_t)dst*42 + 3] - atom14[(size_t)src*42 + 3]);
    float dy = SCALE * (atom14[(size_t)dst*42 + 4] - atom14[(size_t)src*42 + 4]);
    float dz = SCALE * (atom14[(size_t)dst*42 + 5] - atom14[(size_t)src*42 + 5]);
    float dca = __builtin_amdgcn_sqrtf(dx*dx + dy*dy + dz*dz);

    const float sigma_inv = (float)NUM_RBF / 20.0f;    // 1/sigma, sigma = 20/64
    const float nlog2e    = -1.44269504088896340736f;  // -log2(e)
    #pragma unroll
    for (int k = lane; k < NUM_RBF; k += 32) {
        float mu = 20.0f * (float)k / 63.0f;           // linspace(0,20,64)
        float t  = (dca - mu) * sigma_inv;
        row[43 + k] = __builtin_amdgcn_exp2f(t * t * nlog2e);  // exp(-t^2)
    }

    if (lane < 14) {
        row[107 + lane*3 + 0] = Am.x;                  // src local (no EPS)
        row[107 + lane*3 + 1] = Am.y;
        row[107 + lane*3 + 2] = Am.z;
        row[149 + lane*3 + 0] = b0;                    // dst-in-src local
        row[149 + lane*3 + 1] = b1;
        row[149 + lane*3 + 2] = b2;
    }

    // ---- coalesced flush -------------------------------------------------
    float* op = out + (size_t)e * NCOLS;
    #pragma unroll 4
    for (int c = lane; c < NCOLS; c += 32) op[c] = row[c];
}

// ---------------------------------------------------------------------------
extern "C" void kernel_launch(void* const* d_in, const int* in_sizes, int n_in,
                              void* d_out, int out_size, void* d_ws, size_t ws_size,
                              hipStream_t stream)
{
    const float* atom14 = (const float*)d_in[0];   // (N,14,3)
    const float* rot    = (const float*)d_in[1];   // (N,3,3)
    const float* trans  = (const float*)d_in[2];   // (N,3)
    const float* logits = (const float*)d_in[3];   // (N,21)
    const int*   eidx   = (const int*)  d_in[4];   // (2,E)

    int N = in_sizes[0] / 42;
    int E = in_sizes[4] / 2;

    float4* nodeA  = (float4*)d_ws;                               // N*16 float4
    int*    seqArr = (int*)((char*)d_ws + (size_t)N * 16 * sizeof(float4));

    node_pre_kernel<<<(N + 255) / 256, 256, 0, stream>>>(
        atom14, rot, trans, logits, nodeA, seqArr, N);

    int blocks = (E + EPB - 1) / EPB;
    edge_kernel<<<blocks, 256, 0, stream>>>(
        atom14, trans, eidx, nodeA, seqArr, rot, (float*)d_out, E);
}